// MASMG_67877663146453
// MI455X (gfx1250) — compile-verified
//
#include <hip/hip_runtime.h>
#include <cmath>
#include <cstddef>

#define N_NODES 50000
#define N_EDGES 400000
#define FIN     128
#define HID     64
#define NLAYERS 3
#define NGRAPHS 512
#define HEADS   4
#define QKV     (HEADS * HID)   // 256

typedef __attribute__((ext_vector_type(2))) float v2f;
typedef __attribute__((ext_vector_type(8))) float v8f;

// IEEE-ordering float atomic max via integer atomics (works for +/- values).
__device__ __forceinline__ void atomicMaxF32(float* addr, float val) {
    if (val >= 0.0f)
        atomicMax((int*)addr, __float_as_int(val));
    else
        atomicMin((unsigned int*)addr, __float_as_uint(val));
}

// ---------------------------------------------------------------------------
// C[M,Nc] = A[M,K] @ W[K,Nc] + bias  (fp32, WMMA 16x16x4_f32, wave per tile)
// A-frag (16x4):  lane l16(+16*half): v0=A[l16, k+2*half], v1=A[l16, k+2*half+1]
// B-frag (4x16):  lane l16(+16*half): v0=W[k+2*half, c], v1=W[k+2*half+1, c]
// C/D (16x16):    lane l16(+16*half): vr = C[r + 8*half, l16]
// ---------------------------------------------------------------------------
__global__ void gemm_bias_wmma(const float* __restrict__ A,
                               const float* __restrict__ W,
                               const float* __restrict__ bias,
                               float* __restrict__ C,
                               int M, int K, int Nc) {
    const int wave = threadIdx.x >> 5;
    const int lane = threadIdx.x & 31;
    const int tilesX = Nc >> 4;
    const int tilesY = M >> 4;
    int tile = blockIdx.x * (blockDim.x >> 5) + wave;
    if (tile >= tilesX * tilesY) return;
    int ty = tile / tilesX;
    int tx = tile - ty * tilesX;
    int row0 = ty << 4, col0 = tx << 4;
    int half = lane >> 4, l16 = lane & 15;

    v8f acc = {};
    const float* ap = A + (size_t)(row0 + l16) * K + 2 * half;
    const float* bp = W + (size_t)(2 * half) * Nc + col0 + l16;
    for (int k = 0; k < K; k += 4) {
        v2f a, b;
        a.x = ap[0];  a.y = ap[1];
        b.x = bp[0];  b.y = bp[Nc];
        ap += 4;
        bp += (size_t)4 * Nc;
        acc = __builtin_amdgcn_wmma_f32_16x16x4_f32(
            /*neg_a=*/false, a, /*neg_b=*/false, b,
            /*c_mod=*/(short)0, acc, /*reuse_a=*/false, /*reuse_b=*/false);
    }
    float bb = bias[col0 + l16];
#pragma unroll
    for (int r = 0; r < 8; ++r) {
        int row = row0 + r + 8 * half;
        C[(size_t)row * Nc + col0 + l16] = acc[r] + bb;
    }
}

// ---------------------------------------------------------------------------
// Wave per edge: logits[e,h] = (q[dst,h,:]·k[src,h,:]) / 8 ; atomicMax into m.
// Lane l handles channels [8l, 8l+8); head h = l>>3; 8-lane shuffle reduction.
// ---------------------------------------------------------------------------
__global__ void edge_logits_kernel(const float* __restrict__ q,
                                   const float* __restrict__ k,
                                   const int* __restrict__ src,
                                   const int* __restrict__ dst,
                                   float* __restrict__ logits,
                                   float* __restrict__ m) {
    int e = blockIdx.x * (blockDim.x >> 5) + (threadIdx.x >> 5);
    if (e >= N_EDGES) return;
    int lane = threadIdx.x & 31;
    int s = src[e], d = dst[e];
    const float4* qp = (const float4*)(q + (size_t)d * QKV) + lane * 2;
    const float4* kp = (const float4*)(k + (size_t)s * QKV) + lane * 2;
    float4 q0 = qp[0], q1 = qp[1];
    float4 k0 = kp[0], k1 = kp[1];
    float p = q0.x * k0.x + q0.y * k0.y + q0.z * k0.z + q0.w * k0.w
            + q1.x * k1.x + q1.y * k1.y + q1.z * k1.z + q1.w * k1.w;
    p += __shfl_xor(p, 1, 32);
    p += __shfl_xor(p, 2, 32);
    p += __shfl_xor(p, 4, 32);
    if ((lane & 7) == 0) {
        int h = lane >> 3;
        float lg = p * 0.125f;  // 1/sqrt(64)
        logits[(size_t)e * HEADS + h] = lg;
        atomicMaxF32(&m[(size_t)d * HEADS + h], lg);
    }
}

// exp(logit - m[dst]) in place; accumulate segment denominator.
__global__ void edge_exp_kernel(float* __restrict__ logits,
                                const int* __restrict__ dst,
                                const float* __restrict__ m,
                                float* __restrict__ denom) {
    int idx = blockIdx.x * blockDim.x + threadIdx.x;
    if (idx >= N_EDGES * HEADS) return;
    int e = idx >> 2, h = idx & 3;
    int d = dst[e];
    float w = expf(logits[idx] - m[(size_t)d * HEADS + h]);
    logits[idx] = w;
    atomicAdd(&denom[(size_t)d * HEADS + h], w);
}

// Wave per edge: agg[dst,h,:] += p * v[src,h,:]  (p = expw / denom[dst]).
__global__ void edge_agg_kernel(const float* __restrict__ v,
                                const float* __restrict__ expw,
                                const float* __restrict__ denom,
                                const int* __restrict__ src,
                                const int* __restrict__ dst,
                                float* __restrict__ agg) {
    int e = blockIdx.x * (blockDim.x >> 5) + (threadIdx.x >> 5);
    if (e >= N_EDGES) return;
    int lane = threadIdx.x & 31;
    int s = src[e], d = dst[e];
    int h = lane >> 3;
    float p = expw[(size_t)e * HEADS + h] / denom[(size_t)d * HEADS + h];
    const float4* vp = (const float4*)(v + (size_t)s * QKV) + lane * 2;
    float4 v0 = vp[0], v1 = vp[1];
    float* ap = agg + (size_t)d * QKV + lane * 8;
    atomicAdd(ap + 0, p * v0.x); atomicAdd(ap + 1, p * v0.y);
    atomicAdd(ap + 2, p * v0.z); atomicAdd(ap + 3, p * v0.w);
    atomicAdd(ap + 4, p * v1.x); atomicAdd(ap + 5, p * v1.y);
    atomicAdd(ap + 6, p * v1.z); atomicAdd(ap + 7, p * v1.w);
}

// h = relu(mean over heads of agg + skip)   (in-place over h)
__global__ void finalize_kernel(const float* __restrict__ agg,
                                const float* __restrict__ skip,
                                float* __restrict__ h) {
    int idx = blockIdx.x * blockDim.x + threadIdx.x;
    if (idx >= N_NODES * HID) return;
    int node = idx >> 6, c = idx & 63;
    const float* a = agg + (size_t)node * QKV + c;
    float s = 0.25f * (a[0] + a[64] + a[128] + a[192]);
    h[idx] = fmaxf(s + skip[idx], 0.0f);
}

__global__ void fill_kernel(float* __restrict__ p, float val, int n) {
    int idx = blockIdx.x * blockDim.x + threadIdx.x;
    if (idx < n) p[idx] = val;
}

// global add pool: out[batch[n], c] += h[n, c]
__global__ void pool_kernel(const float* __restrict__ h,
                            const int* __restrict__ batch,
                            float* __restrict__ out) {
    int idx = blockIdx.x * blockDim.x + threadIdx.x;
    if (idx >= N_NODES * HID) return;
    int node = idx >> 6, c = idx & 63;
    atomicAdd(&out[(size_t)batch[node] * HID + c], h[idx]);
}

extern "C" void kernel_launch(void* const* d_in, const int* in_sizes, int n_in,
                              void* d_out, int out_size, void* d_ws, size_t ws_size,
                              hipStream_t stream) {
    const float* x      = (const float*)d_in[0];
    const int*   edge   = (const int*)d_in[1];
    const int*   batch  = (const int*)d_in[2];
    const float* lin0_w = (const float*)d_in[3];
    const float* lin0_b = (const float*)d_in[4];
    const float* Wq     = (const float*)d_in[5];
    const float* bq     = (const float*)d_in[6];
    const float* Wk     = (const float*)d_in[7];
    const float* bk     = (const float*)d_in[8];
    const float* Wv     = (const float*)d_in[9];
    const float* bv     = (const float*)d_in[10];
    const float* Ws     = (const float*)d_in[11];
    const float* bs     = (const float*)d_in[12];
    float* out = (float*)d_out;
    (void)in_sizes; (void)n_in; (void)out_size; (void)ws_size;

    const int* src = edge;
    const int* dst = edge + N_EDGES;

    char* ws = (char*)d_ws;
    size_t off = 0;
    auto walloc = [&](size_t bytes) -> float* {
        float* p = (float*)(ws + off);
        off += (bytes + 255) & ~(size_t)255;
        return p;
    };
    float* h    = walloc((size_t)N_NODES * HID * 4);
    float* q    = walloc((size_t)N_NODES * QKV * 4);
    float* kbuf = walloc((size_t)N_NODES * QKV * 4);
    float* vbuf = walloc((size_t)N_NODES * QKV * 4);
    float* skip = walloc((size_t)N_NODES * HID * 4);
    float* agg  = walloc((size_t)N_NODES * QKV * 4);
    float* ew   = walloc((size_t)N_EDGES * HEADS * 4);
    float* mmax = walloc((size_t)N_NODES * HEADS * 4);
    float* den  = walloc((size_t)N_NODES * HEADS * 4);

    const int TB = 256;  // 8 waves/block
    auto gemm = [&](const float* A, const float* Wt, const float* b, float* C,
                    int M, int K, int Nc) {
        int tiles  = (M >> 4) * (Nc >> 4);
        int blocks = (tiles + 7) / 8;
        gemm_bias_wmma<<<blocks, TB, 0, stream>>>(A, Wt, b, C, M, K, Nc);
    };

    // h = x @ lin0_w + lin0_b
    gemm(x, lin0_w, lin0_b, h, N_NODES, FIN, HID);

    const int edgeWaveBlocks = (N_EDGES + 7) / 8;
    for (int l = 0; l < NLAYERS; ++l) {
        const float* Wq_l = Wq + (size_t)l * HID * QKV;
        const float* bq_l = bq + (size_t)l * QKV;
        const float* Wk_l = Wk + (size_t)l * HID * QKV;
        const float* bk_l = bk + (size_t)l * QKV;
        const float* Wv_l = Wv + (size_t)l * HID * QKV;
        const float* bv_l = bv + (size_t)l * QKV;
        const float* Ws_l = Ws + (size_t)l * HID * HID;
        const float* bs_l = bs + (size_t)l * HID;

        gemm(h, Wq_l, bq_l, q,    N_NODES, HID, QKV);
        gemm(h, Wk_l, bk_l, kbuf, N_NODES, HID, QKV);
        gemm(h, Wv_l, bv_l, vbuf, N_NODES, HID, QKV);
        gemm(h, Ws_l, bs_l, skip, N_NODES, HID, HID);

        fill_kernel<<<(N_NODES * HEADS + TB - 1) / TB, TB, 0, stream>>>(
            mmax, -INFINITY, N_NODES * HEADS);
        fill_kernel<<<(N_NODES * HEADS + TB - 1) / TB, TB, 0, stream>>>(
            den, 0.0f, N_NODES * HEADS);
        fill_kernel<<<(N_NODES * QKV + TB - 1) / TB, TB, 0, stream>>>(
            agg, 0.0f, N_NODES * QKV);

        edge_logits_kernel<<<edgeWaveBlocks, TB, 0, stream>>>(q, kbuf, src, dst, ew, mmax);
        edge_exp_kernel<<<(N_EDGES * HEADS + TB - 1) / TB, TB, 0, stream>>>(ew, dst, mmax, den);
        edge_agg_kernel<<<edgeWaveBlocks, TB, 0, stream>>>(vbuf, ew, den, src, dst, agg);
        finalize_kernel<<<(N_NODES * HID + TB - 1) / TB, TB, 0, stream>>>(agg, skip, h);
    }

    fill_kernel<<<(NGRAPHS * HID + TB - 1) / TB, TB, 0, stream>>>(out, 0.0f, NGRAPHS * HID);
    pool_kernel<<<(N_NODES * HID + TB - 1) / TB, TB, 0, stream>>>(h, batch, out);
}